// VectorQuantizer_32109175504939
// MI455X (gfx1250) — compile-verified
//
#include <hip/hip_runtime.h>
#include <hip/hip_bf16.h>

// VQ-VAE vector quantizer for MI455X (gfx1250, wave32).
//   x   : [65536, 64] f32   (latents reshaped)
//   emb : [1024, 64]  f32
//   out : [65536, 64] f32 quantized (= emb[argmin]) followed by 1 f32 loss.
//
// dist[n,k] = ||e_k||^2 - 2 * (x_n . e_k)   (||x||^2 dropped: row-constant)
// Matmul via V_WMMA_F32_16X16X4_F32; A (x tile) resident in VGPRs; B (emb)
// double-buffered across K chunks so loads overlap WMMA instead of the
// load->s_wait_loadcnt 0->wmma serialization the single-buffer version got.

typedef __attribute__((ext_vector_type(2))) float v2f;
typedef __attribute__((ext_vector_type(8))) float v8f;

#define VQ_D          64
#define VQ_K          1024
#define VQ_N          65536
#define ROWS_PER_WAVE 32      // two 16-row WMMA tiles per wave
#define WAVES_PER_BLK 4
#define ROWS_PER_BLK  (ROWS_PER_WAVE * WAVES_PER_BLK)   // 128
#define NBLOCKS       (VQ_N / ROWS_PER_BLK)             // 512
#define NWAVES        (NBLOCKS * WAVES_PER_BLK)         // 2048
#define KCHUNKS       (VQ_K / 16)                       // 64

__global__ __launch_bounds__(128) void vq_main_kernel(
    const float* __restrict__ x, const float* __restrict__ emb,
    float* __restrict__ out, float* __restrict__ partial)
{
    __shared__ int s_idx[WAVES_PER_BLK][ROWS_PER_WAVE];

    const int lane = threadIdx.x & 31;
    const int wave = threadIdx.x >> 5;
    const int half = lane >> 4;        // 0 or 1 (lane group)
    const int lrow = lane & 15;        // row/code within a 16-tile
    const int doff = 2 * half;         // per-lane d offset inside 4-wide K step
    const int rowBase = (blockIdx.x * WAVES_PER_BLK + wave) * ROWS_PER_WAVE;

    // ---- Load A fragments: two 16x64 x-tiles, ISA 16x4 f32 A layout.
    // lane L, step kk holds x[row = L%16][d = 4*kk + 2*(L/16) + {0,1}]
    v2f a[2][16];
#pragma unroll
    for (int t = 0; t < 2; ++t) {
        const float* xr = x + (size_t)(rowBase + t * 16 + lrow) * VQ_D + doff;
#pragma unroll
        for (int kk = 0; kk < 16; ++kk)
            a[t][kk] = *(const v2f*)(xr + 4 * kk);
    }

    float bestd[2][8];
    int   besti[2][8];
#pragma unroll
    for (int t = 0; t < 2; ++t)
#pragma unroll
        for (int r = 0; r < 8; ++r) { bestd[t][r] = 3.402823466e+38f; besti[t][r] = 0; }

    // B fragment loader: same per-lane layout as A (code = lane%16, d-half by lane/16)
    auto loadB = [&](int c, v2f* b) {
        const float* er = emb + (size_t)(c * 16 + lrow) * VQ_D + doff;
#pragma unroll
        for (int kk = 0; kk < 16; ++kk)
            b[kk] = *(const v2f*)(er + 4 * kk);
    };

    // One 16-code chunk of distance work: 32 chained f32 WMMAs + argmin update
    auto compute = [&](int c, const v2f* b) {
        v8f acc0 = {};
        v8f acc1 = {};
        float en = 0.0f;   // partial ||e_code||^2 over this lane's 32 d-values
#pragma unroll
        for (int kk = 0; kk < 16; ++kk) {
            en = fmaf(b[kk].x, b[kk].x, en);
            en = fmaf(b[kk].y, b[kk].y, en);
            acc0 = __builtin_amdgcn_wmma_f32_16x16x4_f32(
                false, a[0][kk], false, b[kk], (short)0, acc0, false, false);
            acc1 = __builtin_amdgcn_wmma_f32_16x16x4_f32(
                false, a[1][kk], false, b[kk], (short)0, acc1, false, false);
        }
        // lanes L and L^16 hold the same code, complementary d-halves
        en += __shfl_xor(en, 16, 32);

        const int code = c * 16 + lrow;   // C/D col N = lane%16
#pragma unroll
        for (int r = 0; r < 8; ++r) {
            float d0 = fmaf(-2.0f, acc0[r], en);
            if (d0 < bestd[0][r]) { bestd[0][r] = d0; besti[0][r] = code; }
            float d1 = fmaf(-2.0f, acc1[r], en);
            if (d1 < bestd[1][r]) { bestd[1][r] = d1; besti[1][r] = code; }
        }
    };

    // ---- K loop: 64 chunks of 16 codes, software-pipelined with two B buffers
    v2f bA[16], bB[16];
    loadB(0, bA);
    for (int c = 0; c < KCHUNKS; c += 2) {
        loadB(c + 1, bB);                    // prefetch odd chunk
        compute(c, bA);                      // compute even chunk
        if (c + 2 < KCHUNKS) loadB(c + 2, bA);  // prefetch next even chunk
        compute(c + 1, bB);                  // compute odd chunk
    }

    // ---- Butterfly argmin across the 16 lanes sharing each row (masks 1..8),
    // tie-break on smaller index to match jnp.argmin first-min semantics.
#pragma unroll
    for (int m = 1; m <= 8; m <<= 1) {
#pragma unroll
        for (int t = 0; t < 2; ++t)
#pragma unroll
            for (int r = 0; r < 8; ++r) {
                float od = __shfl_xor(bestd[t][r], m, 32);
                int   oi = __shfl_xor(besti[t][r], m, 32);
                if (od < bestd[t][r] || (od == bestd[t][r] && oi < besti[t][r])) {
                    bestd[t][r] = od; besti[t][r] = oi;
                }
            }
    }

    // ---- Publish indices: reg r of lanes 0-15 -> row r; lanes 16-31 -> row r+8
    if (lrow == 0) {
#pragma unroll
        for (int t = 0; t < 2; ++t)
#pragma unroll
            for (int r = 0; r < 8; ++r)
                s_idx[wave][t * 16 + half * 8 + r] = besti[t][r];
    }
    __syncthreads();

    // ---- Gather codebook rows, write quantized output, accumulate loss.
    // Each lane covers exactly its resident x values (row lane%16, its d-half).
    float loss = 0.0f;
#pragma unroll
    for (int t = 0; t < 2; ++t) {
        const int idx = s_idx[wave][t * 16 + lrow];
        const float* er = emb + (size_t)idx * VQ_D + doff;
        float* orow = out + (size_t)(rowBase + t * 16 + lrow) * VQ_D + doff;
#pragma unroll
        for (int kk = 0; kk < 16; ++kk) {
            v2f e2 = *(const v2f*)(er + 4 * kk);
            *(v2f*)(orow + 4 * kk) = e2;
            float dx = e2.x - a[t][kk].x;
            float dy = e2.y - a[t][kk].y;
            loss = fmaf(dx, dx, loss);
            loss = fmaf(dy, dy, loss);
        }
    }
    // deterministic full-wave sum
#pragma unroll
    for (int m = 1; m < 32; m <<= 1) loss += __shfl_xor(loss, m, 32);
    if (lane == 0) partial[blockIdx.x * WAVES_PER_BLK + wave] = loss;
}

__global__ __launch_bounds__(256) void vq_reduce_kernel(
    const float* __restrict__ partial, float* __restrict__ out_loss)
{
    __shared__ float s[256];
    float v = 0.0f;
    for (int i = threadIdx.x; i < NWAVES; i += 256) v += partial[i];  // fixed order
    s[threadIdx.x] = v;
    __syncthreads();
    for (int off = 128; off > 0; off >>= 1) {
        if ((int)threadIdx.x < off) s[threadIdx.x] += s[threadIdx.x + off];
        __syncthreads();
    }
    if (threadIdx.x == 0)
        *out_loss = s[0] * (1.25f / 256.0f);   // (1+beta) * sum / B
}

extern "C" void kernel_launch(void* const* d_in, const int* in_sizes, int n_in,
                              void* d_out, int out_size, void* d_ws, size_t ws_size,
                              hipStream_t stream) {
    (void)in_sizes; (void)n_in; (void)out_size; (void)ws_size;
    const float* latents = (const float*)d_in[0];   // [256,16384] == [65536,64]
    const float* emb     = (const float*)d_in[1];   // [1024,64]
    float* out  = (float*)d_out;                    // 65536*64 quantized + 1 loss
    float* part = (float*)d_ws;                     // 2048 per-wave partials

    vq_main_kernel<<<NBLOCKS, 128, 0, stream>>>(latents, emb, out, part);
    vq_reduce_kernel<<<1, 256, 0, stream>>>(part, out + (size_t)VQ_N * VQ_D);
}